// FEA_4956392259740
// MI455X (gfx1250) — compile-verified
//
#include <hip/hip_runtime.h>

// ---------------------------------------------------------------------------
// Problem constants (B,C,H,W fixed by the reference)
// ---------------------------------------------------------------------------
#define BATCH 2
#define CIN   3
#define OC    2
#define Hs    96
#define Ws    96
#define NPIX  (Hs * Ws)            // 9216
#define NTILE 6                    // 96/16 tiles per dim
#define NBLK  (BATCH * NTILE * NTILE)  // 72 conv blocks
#define EPSV  1e-5f
#define INV_SQRT_C 0.57735026918962576f  // 1/sqrt(3)

// workspace layout (floats)
#define WS_G    0     // 18 floats: G[b][c'][c] (scaled by 1/sqrt(C))
#define WS_PART 32    // 288 floats: per-block {sum0,sq0,sum1,sq1}
#define WS_SB   352   // 4 floats: {scale0,bias0,scale1,bias1}

typedef __attribute__((ext_vector_type(2))) float v2f;
typedef __attribute__((ext_vector_type(8))) float v8f;

// ---------------------------------------------------------------------------
// Kernel 1: per-batch 3x3 Gram matrix  G[c'][c] = (1/sqrt(C)) * sum_m k[c',m] q[c,m]
// via V_WMMA_F32_16X16X4_F32 (full fp32 accumulation).
// One block per batch, 8 waves, each wave reduces a 1152-pixel slice.
// Divergence-free: lanes beyond channel 2 load clamped (valid) data; the
// resulting garbage rows/cols of D are never read (matmul rows/cols are
// independent), so no zero-padding or EXEC masking is needed in the loop.
// ---------------------------------------------------------------------------
__global__ __launch_bounds__(256) void gram_kernel(const float* __restrict__ q,   // X_tnext
                                                   const float* __restrict__ kh,  // X_hat_tnext
                                                   float* __restrict__ ws)
{
    const int b    = blockIdx.x;
    const int tid  = threadIdx.x;
    const int wave = tid >> 5;
    const int lane = tid & 31;
    const int row  = lane & 15;           // A: M index (=c'), B: N index (=c)
    const int ksub = (lane >> 4) * 2;     // K sub-offset {0,2} per lane half
    const int rowc = (row < CIN) ? row : 0;  // clamp: dead rows/cols are discarded

    const float* aPtr = kh + (b * CIN + rowc) * NPIX;  // A rows = channels of X_hat
    const float* bPtr = q  + (b * CIN + rowc) * NPIX;  // B cols = channels of X_tnext

    const int perWave = NPIX / 8;         // 1152
    const int p0 = wave * perWave;
    const int p1 = p0 + perWave;

    v8f acc = {};
    #pragma unroll 4
    for (int p = p0; p < p1; p += 4) {
        // A 16x4 f32 layout: lane(0..15) VGPR0=K0, VGPR1=K1; lane(16..31) K2,K3.
        // B 4x16 f32 layout mirrors A: col = lane&15, K split by lane half.
        // index = rowc*9216 + p + ksub is even -> 8B-aligned global_load_b64.
        const v2f a  = *(const v2f*)(aPtr + p + ksub);
        const v2f bm = *(const v2f*)(bPtr + p + ksub);
        acc = __builtin_amdgcn_wmma_f32_16x16x4_f32(
            /*neg_a=*/false, a, /*neg_b=*/false, bm,
            /*c_mod=*/(short)0, acc, /*reuse_a=*/false, /*reuse_b=*/false);
    }

    // D layout: lane n (0..15) = column n, acc[m] = row m (m<8). Useful: rows 0..2, cols 0..2.
    __shared__ float sPart[8][9];
    if (lane < CIN) {
        sPart[wave][0 * 3 + lane] = acc[0];
        sPart[wave][1 * 3 + lane] = acc[1];
        sPart[wave][2 * 3 + lane] = acc[2];
    }
    __syncthreads();
    if (tid < 9) {
        float s = 0.f;
        #pragma unroll
        for (int w = 0; w < 8; ++w) s += sPart[w][tid];
        ws[WS_G + b * 9 + tid] = s * INV_SQRT_C;
    }
}

// ---------------------------------------------------------------------------
// Shared conv helper: agg = G^T * q pointwise (fused into the halo-tile LDS
// load), then 3x3 conv (3->2 channels) for one 16x16 output tile.
// ---------------------------------------------------------------------------
__device__ __forceinline__ void conv_tile(const float* __restrict__ X,
                                          const float* __restrict__ ws,
                                          const float* __restrict__ cw,
                                          const float* __restrict__ cb,
                                          float (*sAgg)[18][19],
                                          float& y0, float& y1,
                                          int& bOut, int& hOut, int& wOut)
{
    const int tid = threadIdx.x;
    const int bx  = blockIdx.x;
    const int b   = bx / (NTILE * NTILE);
    const int t   = bx % (NTILE * NTILE);
    const int th  = (t / NTILE) * 16;
    const int tw  = (t % NTILE) * 16;

    float G[9];
    #pragma unroll
    for (int i = 0; i < 9; ++i) G[i] = ws[WS_G + b * 9 + i];

    const float* Xb = X + b * CIN * NPIX;
    __builtin_prefetch(Xb + th * Ws + tw, 0, 0);   // global_prefetch_b8

    // halo tile: 18x18, 3 agg channels; agg[c] = sum_{c'} q[c'] * G[c'*3+c]
    for (int i = tid; i < 18 * 18; i += 256) {
        const int py = i / 18, px = i % 18;
        const int h = th + py - 1, w = tw + px - 1;
        float q0 = 0.f, q1 = 0.f, q2 = 0.f;
        if (h >= 0 && h < Hs && w >= 0 && w < Ws) {
            const int m = h * Ws + w;
            q0 = Xb[m];
            q1 = Xb[NPIX + m];
            q2 = Xb[2 * NPIX + m];
        }
        sAgg[0][py][px] = q0 * G[0] + q1 * G[3] + q2 * G[6];
        sAgg[1][py][px] = q0 * G[1] + q1 * G[4] + q2 * G[7];
        sAgg[2][py][px] = q0 * G[2] + q1 * G[5] + q2 * G[8];
    }
    __syncthreads();

    const int ty = tid / 16, tx = tid % 16;
    float a0 = cb[0], a1 = cb[1];
    #pragma unroll
    for (int ic = 0; ic < CIN; ++ic) {
        #pragma unroll
        for (int kh = 0; kh < 3; ++kh) {
            #pragma unroll
            for (int kw = 0; kw < 3; ++kw) {
                const float v = sAgg[ic][ty + kh][tx + kw];
                a0 += cw[(0 * CIN + ic) * 9 + kh * 3 + kw] * v;
                a1 += cw[(1 * CIN + ic) * 9 + kh * 3 + kw] * v;
            }
        }
    }
    y0 = a0; y1 = a1;
    bOut = b; hOut = th + ty; wOut = tw + tx;
}

// ---------------------------------------------------------------------------
// Kernel 2: conv pass 1 — deterministic per-block partial sums for BN stats.
// ---------------------------------------------------------------------------
__global__ __launch_bounds__(256) void conv_stats_kernel(const float* __restrict__ X,
                                                         const float* __restrict__ cw,
                                                         const float* __restrict__ cb,
                                                         float* __restrict__ ws)
{
    __shared__ float sAgg[CIN][18][19];
    __shared__ float4 red[256];

    float y0, y1; int b, h, w;
    conv_tile(X, ws, cw, cb, sAgg, y0, y1, b, h, w);

    const int tid = threadIdx.x;
    red[tid] = make_float4(y0, y0 * y0, y1, y1 * y1);
    __syncthreads();
    #pragma unroll
    for (int s = 128; s > 0; s >>= 1) {
        if (tid < s) {
            red[tid].x += red[tid + s].x;
            red[tid].y += red[tid + s].y;
            red[tid].z += red[tid + s].z;
            red[tid].w += red[tid + s].w;
        }
        __syncthreads();
    }
    if (tid == 0) {
        ws[WS_PART + blockIdx.x * 4 + 0] = red[0].x;
        ws[WS_PART + blockIdx.x * 4 + 1] = red[0].y;
        ws[WS_PART + blockIdx.x * 4 + 2] = red[0].z;
        ws[WS_PART + blockIdx.x * 4 + 3] = red[0].w;
    }
}

// ---------------------------------------------------------------------------
// Kernel 3: fold 72 block-partials -> per-channel BN scale/bias.
// ---------------------------------------------------------------------------
__global__ __launch_bounds__(32) void stats_finalize_kernel(const float* __restrict__ gamma,
                                                            const float* __restrict__ beta,
                                                            float* __restrict__ ws)
{
    __shared__ float col[4];
    const int tid = threadIdx.x;
    if (tid < 4) {
        float s = 0.f;
        for (int i = 0; i < NBLK; ++i) s += ws[WS_PART + i * 4 + tid];
        col[tid] = s;
    }
    __syncthreads();
    if (tid < OC) {
        const float cnt  = (float)(BATCH * NPIX);       // 18432 per channel
        const float mean = col[tid * 2 + 0] / cnt;
        const float var  = col[tid * 2 + 1] / cnt - mean * mean;   // biased var
        const float scl  = gamma[tid] * rsqrtf(var + EPSV);
        ws[WS_SB + tid * 2 + 0] = scl;
        ws[WS_SB + tid * 2 + 1] = beta[tid] - mean * scl;
    }
}

// ---------------------------------------------------------------------------
// Kernel 4: conv pass 2 — recompute y, apply BN affine + LeakyReLU(0.1).
// ---------------------------------------------------------------------------
__global__ __launch_bounds__(256) void conv_norm_kernel(const float* __restrict__ X,
                                                        const float* __restrict__ cw,
                                                        const float* __restrict__ cb,
                                                        const float* __restrict__ ws,
                                                        float* __restrict__ out)
{
    __shared__ float sAgg[CIN][18][19];

    float y0, y1; int b, h, w;
    conv_tile(X, ws, cw, cb, sAgg, y0, y1, b, h, w);

    const float s0 = ws[WS_SB + 0], b0 = ws[WS_SB + 1];
    const float s1 = ws[WS_SB + 2], b1 = ws[WS_SB + 3];
    float v0 = y0 * s0 + b0;
    float v1 = y1 * s1 + b1;
    v0 = (v0 >= 0.f) ? v0 : 0.1f * v0;
    v1 = (v1 >= 0.f) ? v1 : 0.1f * v1;

    out[((b * OC + 0) * Hs + h) * Ws + w] = v0;
    out[((b * OC + 1) * Hs + h) * Ws + w] = v1;
}

// ---------------------------------------------------------------------------
// Host launcher
// ---------------------------------------------------------------------------
extern "C" void kernel_launch(void* const* d_in, const int* in_sizes, int n_in,
                              void* d_out, int out_size, void* d_ws, size_t ws_size,
                              hipStream_t stream) {
    const float* X     = (const float*)d_in[0];  // X_tnext      [2,3,96,96]
    const float* Xhat  = (const float*)d_in[1];  // X_hat_tnext  [2,3,96,96]
    const float* convw = (const float*)d_in[2];  // conv_w       [2,3,3,3]
    const float* convb = (const float*)d_in[3];  // conv_b       [2]
    const float* gamma = (const float*)d_in[4];  // gamma        [2]
    const float* beta  = (const float*)d_in[5];  // beta         [2]
    float* out = (float*)d_out;
    float* ws  = (float*)d_ws;
    (void)in_sizes; (void)n_in; (void)out_size; (void)ws_size;

    gram_kernel<<<BATCH, 256, 0, stream>>>(X, Xhat, ws);
    conv_stats_kernel<<<NBLK, 256, 0, stream>>>(X, convw, convb, ws);
    stats_finalize_kernel<<<1, 32, 0, stream>>>(gamma, beta, ws);
    conv_norm_kernel<<<NBLK, 256, 0, stream>>>(X, convw, convb, ws, out);
}